// EmaVectorQuantizer_6390911337159
// MI455X (gfx1250) — compile-verified
//
#include <hip/hip_runtime.h>

// Vector quantization forward for MI455X (gfx1250, wave32).
// dist = |z|^2 - 2 z.c + |c|^2 ; heavy part (z @ codebook^T) done with
// V_WMMA_F32_16X16X4_F32 (f32 matrix path keeps argmin faithful to the f32
// reference). Codebook blocks are staged into double-buffered LDS by the
// Tensor Data Mover (tensor_load_to_lds + s_wait_tensorcnt), with the TDM
// pad feature inserting 4 DWORDs per 128-DWORD row so the staged stride is
// 132 floats -> bank-conflict-free B-fragment reads (all 64 banks hit once).
// Outputs (concatenated): z_q [65536*128], quant_loss [65536],
// commit_loss [65536], indices-as-float [65536].

typedef __attribute__((ext_vector_type(2))) float    v2f;
typedef __attribute__((ext_vector_type(8))) float    v8f;
typedef __attribute__((ext_vector_type(4))) unsigned v4u;
typedef __attribute__((ext_vector_type(4))) int      v4i;
typedef __attribute__((ext_vector_type(8))) int      v8i;

#define NUM_EMBS 1024
#define EMB_DIM  128
#define CB_STRIDE 132  // padded LDS row stride (4-DWORD pad per row)
#define M_TOTAL  (16 * 4096)
#define WAVES_PER_BLOCK 8
#define ROWS_PER_BLOCK  (WAVES_PER_BLOCK * 16)
#define NBLK (NUM_EMBS / 16)  // 64 blocks of 16 codes

#if __has_builtin(__builtin_amdgcn_tensor_load_to_lds_d2) || \
    __has_builtin(__builtin_amdgcn_tensor_load_to_lds)
#define VQ_HAVE_TDM 1
#else
#define VQ_HAVE_TDM 0
#endif

__global__ __launch_bounds__(256) void vq_cnorm_kernel(
    const float* __restrict__ cb, float* __restrict__ cnorm) {
  int c = blockIdx.x * blockDim.x + threadIdx.x;
  if (c < NUM_EMBS) {
    const float4* p = (const float4*)(cb + (size_t)c * EMB_DIM);
    float s = 0.0f;
#pragma unroll
    for (int i = 0; i < EMB_DIM / 4; ++i) {
      float4 v = p[i];
      s += v.x * v.x + v.y * v.y + v.z * v.z + v.w * v.w;
    }
    cnorm[c] = s;
  }
}

#if VQ_HAVE_TDM
// Issue a TDM DMA: 16x128 f32 tile (one 16-code block) global -> LDS with
// a 4-DWORD pad after every 128-DWORD row (LDS stride = 132 floats).
// D# per CDNA5 ISA §8.3/8.4: group0 = {count=1 | lds_addr | global_addr |
// type=2}, group1 = {data_size=4B, pad 4dw/128dw, tensor 128x16, tile
// 128x16, row pitch 128}.
__device__ __forceinline__ void vq_tdm_load_block(const float* gsrc,
                                                  unsigned lds_addr) {
  unsigned long long ga = (unsigned long long)(uintptr_t)gsrc;
  v4u g0;
  g0[0] = 1u;                                              // count=1, user D#
  g0[1] = lds_addr;                                        // LDS byte address
  g0[2] = (unsigned)(ga & 0xffffffffu);                    // global_addr[31:0]
  g0[3] = (unsigned)((ga >> 32) & 0x01ffffffu) | (2u << 30);  // [56:32]|type=2
  v8i g1;
  g1[0] = (int)((2u << 16)     // wg_mask=0, data_size=2 (4 bytes)
              | (1u << 20)     // pad_enable
              | (6u << 22)     // pad_interval: after 1<<6 * 2 = 128 DWORDs
              | (3u << 25));   // pad_amount: 4 DWORDs
  g1[1] = (int)(128u << 16);   // tensor_dim0 = 128 elements (row length)
  g1[2] = (int)(16u << 16);    // tensor_dim0_hi=0 | tensor_dim1 = 16 rows
  g1[3] = (int)(128u << 16);   // tensor_dim1_hi=0 | tile_dim0 = 128
  g1[4] = (int)16u;            // tile_dim1 = 16, tile_dim2 = 0 (2-D tile)
  g1[5] = (int)128u;           // tensor_dim0_stride = 128 (row pitch)
  g1[6] = 0;                   // stride hi / dim1_stride lo (unused for 2-D)
  g1[7] = 0;
#if __has_builtin(__builtin_amdgcn_tensor_load_to_lds_d2)
  __builtin_amdgcn_tensor_load_to_lds_d2(g0, g1, 0);
#else
  // clang-23 / therock-10.0 6-arg form:
  // (uint32x4 g0, int32x8 g1, int32x4 g2, int32x4 g3, int32x8 g4, i32 cpol)
  v4i zz4 = {0, 0, 0, 0};
  v8i zz8 = {0, 0, 0, 0, 0, 0, 0, 0};
  __builtin_amdgcn_tensor_load_to_lds(g0, g1, zz4, zz4, zz8, 0);
#endif
}
#endif  // VQ_HAVE_TDM

__global__ __launch_bounds__(256)
__attribute__((amdgpu_waves_per_eu(2)))  // 8 waves/WGP = 2/SIMD; allow VGPRs
void vq_main_kernel(
    const float* __restrict__ z, const float* __restrict__ cb,
    const float* __restrict__ cnorm, float* __restrict__ out) {
  __shared__ float s_cb[2][16 * CB_STRIDE];     // double-buffered, padded
  __shared__ float s_cn[NUM_EMBS];              // all |c|^2 (4 KB, loaded once)
  __shared__ float s_rn[WAVES_PER_BLOCK][16];   // |z|^2 per row of wave tile

  const int tid  = threadIdx.x;
  const int wave = tid >> 5;
  const int lane = tid & 31;
  const int half = lane >> 4;  // 0: lanes 0-15, 1: lanes 16-31
  const int n16  = lane & 15;

  const int row_base = blockIdx.x * ROWS_PER_BLOCK + wave * 16;
  const int rowA     = row_base + n16;

  // ---- Preload A (16x128 z tile) in WMMA 16x4 f32 layout, 32 k-steps ----
  // lane (half,n16): a[s] = { z[rowA][4s + 2*half], z[rowA][4s + 1 + 2*half] }
  v2f a[32];
  const float* zr = z + (size_t)rowA * EMB_DIM + 2 * half;
#pragma unroll
  for (int s = 0; s < 32; ++s) {
    float2 t = *(const float2*)(zr + 4 * s);
    a[s].x = t.x;
    a[s].y = t.y;
  }

  // ---- |z|^2 per row: each lane has half the row; add halves via shfl ----
  float pn = 0.0f;
#pragma unroll
  for (int s = 0; s < 32; ++s) pn += a[s].x * a[s].x + a[s].y * a[s].y;
  pn += __shfl_xor(pn, 16, 32);
  if (half == 0) s_rn[wave][n16] = pn;

  // ---- stage all code norms once: 1024 floats = 256 x float4 ----
  ((float4*)s_cn)[tid] = ((const float4*)cnorm)[tid];

  // ---- prime buffer 0 with code block 0 ----
#if VQ_HAVE_TDM
  const unsigned lds_cb0 = (unsigned)(uintptr_t)&s_cb[0][0];
  const unsigned lds_cb1 = (unsigned)(uintptr_t)&s_cb[1][0];
  if (wave == 0) vq_tdm_load_block(cb, lds_cb0);
#else
  {
    const float4* src = (const float4*)cb;
#pragma unroll
    for (int i = tid; i < 512; i += 256) {
      int row = i >> 5, col = i & 31;
      *(float4*)(s_cb[0] + row * CB_STRIDE + col * 4) = src[i];
    }
  }
#endif

  float bestv[8];
  int   besti[8];
#pragma unroll
  for (int r = 0; r < 8; ++r) { bestv[r] = 3.4e38f; besti[r] = 0; }

  // ---- sweep 64 blocks of 16 codes, double-buffered ----
#pragma unroll 1
  for (int j = 0; j < NBLK; ++j) {
    const int buf = j & 1;
#if VQ_HAVE_TDM
    if (wave == 0) __builtin_amdgcn_s_wait_tensorcnt(0);  // buf ready
#endif
    __syncthreads();  // all waves past prior reads; staged data visible

    // kick next block's staging into the other buffer (overlaps compute)
    if (j + 1 < NBLK) {
#if VQ_HAVE_TDM
      if (wave == 0)
        vq_tdm_load_block(cb + (size_t)(j + 1) * 16 * EMB_DIM,
                          buf ? lds_cb0 : lds_cb1);
#else
      const float4* src = (const float4*)(cb + (size_t)(j + 1) * 16 * EMB_DIM);
#pragma unroll
      for (int i = tid; i < 512; i += 256) {
        int row = i >> 5, col = i & 31;
        *(float4*)(s_cb[buf ^ 1] + row * CB_STRIDE + col * 4) = src[i];
      }
#endif
    }

    // ---- prefetch all B fragments (one ds clause), then 32 WMMAs ----
    // B layout mirrors A: lane (half,n16): b = { blk[n16][4s+2h], [..+1] }
    // padded stride 132 -> banks (4*n16 + 4s + 2h) % 64: conflict-free.
    v2f bq[32];
    const float* bp = s_cb[buf] + n16 * CB_STRIDE + 2 * half;
#pragma unroll
    for (int s = 0; s < 32; ++s) {
      float2 t = *(const float2*)(bp + 4 * s);
      bq[s].x = t.x;
      bq[s].y = t.y;
    }
#if __has_builtin(__builtin_amdgcn_sched_barrier)
    __builtin_amdgcn_sched_barrier(0);  // keep loads clustered before WMMAs
#endif
    v8f acc = {};
#pragma unroll
    for (int s = 0; s < 32; ++s) {
      acc = __builtin_amdgcn_wmma_f32_16x16x4_f32(
          /*neg_a=*/false, a[s], /*neg_b=*/false, bq[s],
          /*c_mod=*/(short)0, acc, /*reuse_a=*/false, /*reuse_b=*/false);
    }

    // score = |c|^2 - 2*dot ; C tile: VGPR r -> row M=r+8*half, col N=n16
    const float cn   = s_cn[j * 16 + n16];
    const int   code = j * 16 + n16;
#pragma unroll
    for (int r = 0; r < 8; ++r) {
      float sc = cn - 2.0f * acc[r];
      if (sc < bestv[r]) { bestv[r] = sc; besti[r] = code; }
    }
  }

  // ---- argmin across the 16 lanes of each half (ties -> lower index) ----
#pragma unroll
  for (int m = 1; m < 16; m <<= 1) {
#pragma unroll
    for (int r = 0; r < 8; ++r) {
      float ov = __shfl_xor(bestv[r], m, 16);
      int   oi = __shfl_xor(besti[r], m, 16);
      if (ov < bestv[r] || (ov == bestv[r] && oi < besti[r])) {
        bestv[r] = ov;
        besti[r] = oi;
      }
    }
  }

  float* zq = out;
  float* ql = out + (size_t)M_TOTAL * EMB_DIM;
  float* cl = ql + M_TOTAL;
  float* io = cl + M_TOTAL;

  // ---- gather winning codebook rows + write losses/indices ----
#pragma unroll
  for (int r = 0; r < 8; ++r) {
    const int row = row_base + r + 8 * half;
    const int idx = besti[r];  // uniform within each 16-lane half
    const float4* crow = (const float4*)(cb + (size_t)idx * EMB_DIM);
    float4*       orow = (float4*)(zq + (size_t)row * EMB_DIM);
    orow[n16 * 2]     = crow[n16 * 2];
    orow[n16 * 2 + 1] = crow[n16 * 2 + 1];
    if (n16 == 0) {
      float dist = s_rn[wave][r + 8 * half] + bestv[r];
      float loss = dist * (1.0f / 128.0f);  // mean over D of (z_q - z)^2
      ql[row] = loss;
      cl[row] = loss;  // forward values of quant/commit losses are identical
      io[row] = (float)idx;
    }
  }
}

extern "C" void kernel_launch(void* const* d_in, const int* in_sizes, int n_in,
                              void* d_out, int out_size, void* d_ws, size_t ws_size,
                              hipStream_t stream) {
  (void)in_sizes; (void)n_in; (void)out_size; (void)ws_size;
  const float* z  = (const float*)d_in[0];  // (16,4096,128) f32
  const float* cb = (const float*)d_in[1];  // (1024,128) f32
  float* cnorm = (float*)d_ws;              // 1024 floats scratch
  float* out   = (float*)d_out;

  vq_cnorm_kernel<<<dim3((NUM_EMBS + 255) / 256), dim3(256), 0, stream>>>(cb, cnorm);
  vq_main_kernel<<<dim3(M_TOTAL / ROWS_PER_BLOCK), dim3(256), 0, stream>>>(z, cb, cnorm, out);
}